// ASG_12610023981193
// MI455X (gfx1250) — compile-verified
//
#include <hip/hip_runtime.h>

#define T_DIM 2000
#define B_DIM 512
#define C_DIM 29
#define L_DIM 150
#define NEGF  (-1e30f)
#define LOG2E 1.4426950408889634f
#define LN2   0.6931471805599453f

typedef float v2f __attribute__((ext_vector_type(2)));
typedef float v8f __attribute__((ext_vector_type(8)));

// Raw hardware transcendentals (no libm denorm-range fixup):
//   fast_exp2(x): v_exp_f32, correct flush-to-zero for very negative x
//   fast_log2(x): v_log_f32, inputs here are always normal and > 0
__device__ __forceinline__ float fast_exp2(float x) { return __builtin_amdgcn_exp2f(x); }
__device__ __forceinline__ float fast_log2(float x) { return __builtin_amdgcn_logf(x); }

// ---------------------------------------------------------------------------
// Denominator: alpha_t[b,i] = x_t[b,i] + m_b + ln( sum_j E[i,j] * exp(alpha_{t-1}[b,j]-m_b) )
// E = exp(trans) is constant -> per-step GEMM via V_WMMA_F32_16X16X4_F32.
// One wave (32 lanes) per 16-batch tile; LDS holds alpha as [16 b][32 c-padded].
// ---------------------------------------------------------------------------
__global__ __launch_bounds__(32) void asg_den_kernel(
    const float* __restrict__ inputs, const float* __restrict__ transitions,
    float* __restrict__ logden) {
  __shared__ __align__(16) float s_alpha[16 * 32];
  const int lane = threadIdx.x;
  const int half = lane >> 4;   // K-half for A/B operands
  const int ln   = lane & 15;
  const int b0   = blockIdx.x * 16;

  // Constant B-operand: Et[k,n] = exp(trans[1 + n + 16*tile, k]); zero-padded.
  // Layout per 4xK chunk: VGPR0 = rows {k0, k0+2} (lane halves), VGPR1 = {k0+1, k0+3}.
  v2f Bop[2][8];
#pragma unroll
  for (int tile = 0; tile < 2; ++tile) {
#pragma unroll
    for (int ch = 0; ch < 8; ++ch) {
      const int n  = ln + 16 * tile;
      const int k0 = 4 * ch + 2 * half;
      float e0 = 0.0f, e1 = 0.0f;
      if (n < C_DIM) {
        if (k0 < C_DIM)     e0 = fast_exp2(transitions[(1 + n) * C_DIM + k0] * LOG2E);
        if (k0 + 1 < C_DIM) e1 = fast_exp2(transitions[(1 + n) * C_DIM + k0 + 1] * LOG2E);
      }
      Bop[tile][ch] = (v2f){e0, e1};
    }
  }

  // alpha0 = inputs[0,b,:] + transitions[0,:]; pads = NEGF
  {
    const float st = (lane < C_DIM) ? transitions[lane] : 0.0f;
#pragma unroll 4
    for (int b = 0; b < 16; ++b) {
      float v = NEGF;
      if (lane < C_DIM) v = inputs[(size_t)(b0 + b) * C_DIM + lane] + st;
      s_alpha[b * 32 + lane] = v;
    }
  }
  __syncthreads();

  for (int t = 1; t < T_DIM; ++t) {
    const float* __restrict__ xrow = inputs + (size_t)t * (B_DIM * C_DIM);
    if (t + 1 < T_DIM) {  // prefetch next timestep's tile (global_prefetch_b8)
      const float* nxt = xrow + (size_t)(B_DIM * C_DIM) + (size_t)b0 * C_DIM;
      __builtin_prefetch(nxt + lane * 16, 0, 0);
    }

    // A-layout read from LDS: lane = batch, this lane's K pairs
    float2 a2[8];
#pragma unroll
    for (int ch = 0; ch < 8; ++ch) {
      const int k = 4 * ch + 2 * half;
      a2[ch] = *(const float2*)&s_alpha[ln * 32 + k];
    }

    // m_b = max over classes (pads are NEGF, harmless); combine lane halves
    float m = fmaxf(a2[0].x, a2[0].y);
#pragma unroll
    for (int ch = 1; ch < 8; ++ch) m = fmaxf(m, fmaxf(a2[ch].x, a2[ch].y));
    m = fmaxf(m, __shfl_xor(m, 16, 32));

    // u = exp(alpha - m); D = u * Et  (two 16-class N-tiles, 8 K-chunks)
    v8f acc0 = (v8f){0,0,0,0,0,0,0,0};
    v8f acc1 = (v8f){0,0,0,0,0,0,0,0};
#pragma unroll
    for (int ch = 0; ch < 8; ++ch) {
      v2f u;
      u.x = fast_exp2((a2[ch].x - m) * LOG2E);
      u.y = fast_exp2((a2[ch].y - m) * LOG2E);
      acc0 = __builtin_amdgcn_wmma_f32_16x16x4_f32(
          false, u, false, Bop[0][ch], (short)0, acc0, false, false);
      acc1 = __builtin_amdgcn_wmma_f32_16x16x4_f32(
          false, u, false, Bop[1][ch], (short)0, acc1, false, false);
    }

    __syncthreads();  // all reads of old alpha done before overwrite
#pragma unroll
    for (int r = 0; r < 8; ++r) {
      const int   brow = r + 8 * half;            // D: M = r (lanes<16), r+8 (lanes>=16)
      const float mr   = __shfl(m, brow, 32);     // m for this D-row's batch
      const size_t xb  = (size_t)(b0 + brow) * C_DIM;
#pragma unroll
      for (int tile = 0; tile < 2; ++tile) {
        const int   c  = ln + 16 * tile;          // D: N = lane
        const int   cc = (c < C_DIM) ? c : 0;
        const float x  = xrow[xb + cc];
        const float v  = (tile == 0) ? acc0[r] : acc1[r];
        const float av = (c < C_DIM) ? (x + mr + LN2 * fast_log2(v)) : NEGF;
        s_alpha[brow * 32 + c] = av;
      }
    }
    __syncthreads();
  }

  // log_den[b] = logsumexp_c alpha_T[b,c]
  {
    float2 a2[8];
#pragma unroll
    for (int ch = 0; ch < 8; ++ch) {
      const int k = 4 * ch + 2 * half;
      a2[ch] = *(const float2*)&s_alpha[ln * 32 + k];
    }
    float m = fmaxf(a2[0].x, a2[0].y);
#pragma unroll
    for (int ch = 1; ch < 8; ++ch) m = fmaxf(m, fmaxf(a2[ch].x, a2[ch].y));
    m = fmaxf(m, __shfl_xor(m, 16, 32));
    float s = 0.0f;
#pragma unroll
    for (int ch = 0; ch < 8; ++ch)
      s += fast_exp2((a2[ch].x - m) * LOG2E) + fast_exp2((a2[ch].y - m) * LOG2E);
    s += __shfl_xor(s, 16, 32);
    if (lane < 16) logden[b0 + ln] = m + LN2 * fast_log2(s);
  }
}

// ---------------------------------------------------------------------------
// Numerator: banded logaddexp recurrence over L=150 path states.
// One wave per batch; 5 consecutive states per lane (lanes 0..29).
// ---------------------------------------------------------------------------
__global__ __launch_bounds__(256) void asg_num_kernel(
    const float* __restrict__ inputs, const float* __restrict__ transitions,
    const int* __restrict__ targets, float* __restrict__ lognum) {
  const int lane = threadIdx.x & 31;
  const int wave = threadIdx.x >> 5;
  const int b    = blockIdx.x * 8 + wave;
  const bool active = (lane < 30);
  const int l0 = lane * 5;

  int   tgt[5];
  float stay[5], mv[5];
#pragma unroll
  for (int j = 0; j < 5; ++j) {
    const int tg = active ? targets[b * L_DIM + l0 + j] : 0;
    tgt[j]  = tg;
    stay[j] = transitions[(1 + tg) * C_DIM + tg];
  }
  const int prevt4 = __shfl_up(tgt[4], 1, 32);
#pragma unroll
  for (int j = 0; j < 5; ++j) {
    const int l  = l0 + j;
    const int tp = (j == 0) ? prevt4 : tgt[j - 1];
    mv[j] = (l == 0) ? 0.0f : transitions[(1 + tgt[j]) * C_DIM + tp];
  }

  // num0: only l==0 live: start[tgt0] + e_y[0]
  const float* __restrict__ x0 = inputs + (size_t)b * C_DIM;
  float alpha[5];
#pragma unroll
  for (int j = 0; j < 5; ++j) {
    const int l = l0 + j;
    alpha[j] = (l == 0) ? (transitions[tgt[0]] + x0[tgt[0]]) : NEGF;
  }

  const float* __restrict__ x = inputs + (size_t)(B_DIM * C_DIM) + (size_t)b * C_DIM;
  for (int t = 1; t < T_DIM; ++t) {
    float sh0 = __shfl_up(alpha[4], 1, 32);
    if (lane == 0) sh0 = NEGF;
    float e[5];
#pragma unroll
    for (int j = 0; j < 5; ++j) e[j] = x[tgt[j]];  // gather e_y[t,b,l]
    float na[5];
#pragma unroll
    for (int j = 0; j < 5; ++j) {
      const float sh = (j == 0) ? sh0 : alpha[j - 1];
      const float p  = alpha[j] + stay[j];
      const float q  = sh + mv[j];
      const float mx = fmaxf(p, q);
      const float mn = fminf(p, q);
      na[j] = e[j] + mx + LN2 * fast_log2(1.0f + fast_exp2((mn - mx) * LOG2E));
    }
#pragma unroll
    for (int j = 0; j < 5; ++j) alpha[j] = na[j];
    x += B_DIM * C_DIM;
  }
  if (lane == 29) lognum[b] = alpha[4];  // l = 149
}

// ---------------------------------------------------------------------------
// loss = mean(log_den - log_num)
// ---------------------------------------------------------------------------
__global__ __launch_bounds__(256) void asg_reduce_kernel(
    const float* __restrict__ logden, const float* __restrict__ lognum,
    float* __restrict__ out) {
  __shared__ float sm[256];
  const int tid = threadIdx.x;
  float s = 0.0f;
  for (int i = tid; i < B_DIM; i += 256) s += logden[i] - lognum[i];
  sm[tid] = s;
  __syncthreads();
  for (int off = 128; off > 0; off >>= 1) {
    if (tid < off) sm[tid] += sm[tid + off];
    __syncthreads();
  }
  if (tid == 0) out[0] = sm[0] * (1.0f / B_DIM);
}

extern "C" void kernel_launch(void* const* d_in, const int* in_sizes, int n_in,
                              void* d_out, int out_size, void* d_ws, size_t ws_size,
                              hipStream_t stream) {
  const float* inputs      = (const float*)d_in[0];  // (2000, 512, 29) f32
  const float* transitions = (const float*)d_in[1];  // (30, 29) f32
  const int*   targets     = (const int*)d_in[2];    // (512, 150) i32
  float* out    = (float*)d_out;
  float* logden = (float*)d_ws;
  float* lognum = logden + B_DIM;

  asg_den_kernel<<<B_DIM / 16, 32, 0, stream>>>(inputs, transitions, logden);
  asg_num_kernel<<<B_DIM / 8, 256, 0, stream>>>(inputs, transitions, targets, lognum);
  asg_reduce_kernel<<<1, 256, 0, stream>>>(logden, lognum, out);
}